// SoftRasterizer_46351287058959
// MI455X (gfx1250) — compile-verified
//
#include <hip/hip_runtime.h>
#include <hip/hip_bf16.h>

// Soft polygon rasterizer for MI455X (gfx1250).
// dist^2(pixel, segment) is expanded into two 16x16 tiles produced by
// V_WMMA_F32_16X16X4_F32:  D1 = dot(side, rel),  D2 = |rel|^2, using pixel
// feature vector [x, y, x^2+y^2, 1] (K=4) against per-side coefficients.
// Coordinates are re-based per 32-pixel chunk to keep the expansion well
// conditioned. Post-processing (clamp-t, min over 32 sides, sqrt, sigmoid)
// is plain VALU + one shfl_xor(16) reduction.

#define IMG_H 384
#define IMG_W 1280
#define NBATCH 4
#define NVERT 32
#define HW (IMG_H * IMG_W)          // 491520
#define CHUNKS (HW / 32)            // 15360, 32 pixels per chunk (never crosses a row: 32 | 1280)
#define WAVES_PER_BATCH 1024

typedef float v2f __attribute__((ext_vector_type(2)));
typedef float v8f __attribute__((ext_vector_type(8)));

__global__ __launch_bounds__(256) void SoftRasterizer_wmma_kernel(
    const float* __restrict__ poly,   // [4][32][2] (x, y)
    const int*   __restrict__ masks,  // [4][384][1280] (bool -> int per harness contract)
    float*       __restrict__ out)    // [4][384][1280] float32
{
    const int lane = threadIdx.x & 31;
    const int wid  = blockIdx.x * (blockDim.x >> 5) + (threadIdx.x >> 5);
    const int b    = wid >> 10;                 // WAVES_PER_BATCH = 1024
    const int li   = wid & (WAVES_PER_BATCH - 1);
    if (b >= NBATCH) return;                    // uniform per wave

    const int  l15 = lane & 15;
    const bool low = lane < 16;

    // ---- per-lane side features: this lane owns side `lane` ----
    const float* pg = poly + b * (NVERT * 2);
    const float p0x = pg[2 * lane];
    const float p0y = pg[2 * lane + 1];
    const int   ln  = (lane + 1) & (NVERT - 1);
    const float sx  = pg[2 * ln]     - p0x;
    const float sy  = pg[2 * ln + 1] - p0y;
    const float ss  = sx * sx + sy * sy;
    const float iss = 1.0f / (ss + 1e-6f);

    // ---- loop-invariant cross-lane gathers ----
    // A-tile t needs side/prev coords of side (t*16 + l15).
    float sxA[2], syA[2], pxA[2], pyA[2];
#pragma unroll
    for (int t = 0; t < 2; ++t) {
        const int src = t * 16 + l15;
        sxA[t] = __shfl(sx,  src, 32);
        syA[t] = __shfl(sy,  src, 32);
        pxA[t] = __shfl(p0x, src, 32);
        pyA[t] = __shfl(p0y, src, 32);
    }
    // D element (VGPR j, this lane half) corresponds to side t*16 + j + (low?0:8).
    float ssv[2][8], issv[2][8];
    const int hoff = low ? 0 : 8;
#pragma unroll
    for (int t = 0; t < 2; ++t) {
#pragma unroll
        for (int j = 0; j < 8; ++j) {
            const int src = t * 16 + j + hoff;
            ssv[t][j]  = __shfl(ss,  src, 32);
            issv[t][j] = __shfl(iss, src, 32);
        }
    }

    const int baseImg = b * HW;

    for (int c = li; c < CHUNKS; c += WAVES_PER_BATCH) {
        const int   base = c * 32;
        const float ox   = (float)(base % IMG_W);   // chunk origin (row-aligned)
        const float oy   = (float)(base / IMG_W);

        // ---- build A matrices in rebased coordinates ----
        // A layout (16x4 f32): lanes 0-15 hold K0/K1 (VGPR0/1) for M=lane,
        //                      lanes 16-31 hold K2/K3 for M=lane-16.
        // A1 row s = [ sx, sy, 0, -(prev'.side) ]   -> dot(side, rel)
        // A2 row s = [ -2*prev'x, -2*prev'y, 1, |prev'|^2 ] -> |rel|^2
        v2f A1[2], A2[2];
#pragma unroll
        for (int t = 0; t < 2; ++t) {
            const float pxT = pxA[t] - ox;
            const float pyT = pyA[t] - oy;
            const float psT = pxT * sxA[t] + pyT * syA[t];
            const float pqT = pxT * pxT + pyT * pyT;
            A1[t].x = low ? sxA[t] : 0.0f;
            A1[t].y = low ? syA[t] : -psT;
            A2[t].x = low ? (-2.0f * pxT) : 1.0f;
            A2[t].y = low ? (-2.0f * pyT) : pqT;
        }

        float dist[2];
#pragma unroll
        for (int q = 0; q < 2; ++q) {
            // Pixel N = l15 of this tile; rebased coords: x' = q*16+l15, y' = 0.
            const float fx = (float)(q * 16 + l15);
            // B layout (4x16 f32): lanes 0-15 hold K0/K1 = (x', y'),
            //                      lanes 16-31 hold K2/K3 = (x'^2+y'^2, 1).
            v2f Bv;
            Bv.x = low ? fx   : fx * fx;
            Bv.y = low ? 0.0f : 1.0f;

            const v8f zero = {0.f, 0.f, 0.f, 0.f, 0.f, 0.f, 0.f, 0.f};
            // D = A x B + 0  (8 args: neg_a, A, neg_b, B, c_mod, C, reuse_a, reuse_b)
            v8f D1a = __builtin_amdgcn_wmma_f32_16x16x4_f32(false, A1[0], false, Bv, (short)0, zero, false, false);
            v8f D2a = __builtin_amdgcn_wmma_f32_16x16x4_f32(false, A2[0], false, Bv, (short)0, zero, false, false);
            v8f D1b = __builtin_amdgcn_wmma_f32_16x16x4_f32(false, A1[1], false, Bv, (short)0, zero, false, false);
            v8f D2b = __builtin_amdgcn_wmma_f32_16x16x4_f32(false, A2[1], false, Bv, (short)0, zero, false, false);

            float m = 3.4e38f;
#pragma unroll
            for (int j = 0; j < 8; ++j) {
                {
                    const float s = D1a[j], r = D2a[j];
                    float tt = s * issv[0][j];
                    tt = fminf(fmaxf(tt, 0.0f), 1.0f);
                    const float d2 = r - tt * (2.0f * s - tt * ssv[0][j]);
                    m = fminf(m, d2);
                }
                {
                    const float s = D1b[j], r = D2b[j];
                    float tt = s * issv[1][j];
                    tt = fminf(fmaxf(tt, 0.0f), 1.0f);
                    const float d2 = r - tt * (2.0f * s - tt * ssv[1][j]);
                    m = fminf(m, d2);
                }
            }
            // lane half covers sides {j+hoff, 16+j+hoff}; merge halves -> all 32 sides
            m = fminf(m, __shfl_xor(m, 16, 32));
            dist[q] = sqrtf(fmaxf(m, 0.0f));
        }

        // lane l writes pixel base + l (low lanes: tile 0, high lanes: tile 1)
        const float d    = low ? dist[0] : dist[1];
        const int   gp   = baseImg + base + lane;
        const int   mk   = masks[gp];
        const float sdf  = mk ? d : -d;
        out[gp] = 1.0f / (1.0f + __expf(-sdf * 0.1f));   // sigmoid(sdf / 10)
    }
}

extern "C" void kernel_launch(void* const* d_in, const int* in_sizes, int n_in,
                              void* d_out, int out_size, void* d_ws, size_t ws_size,
                              hipStream_t stream) {
    const float* poly  = (const float*)d_in[0];  // [4][32][2] float32
    const int*   masks = (const int*)d_in[1];    // [4][384][1280] bool -> int
    float*       out   = (float*)d_out;          // [4][384][1280] float32

    dim3 grid(512);   // 512 blocks * 8 waves = 4096 waves = 1024 per batch
    dim3 block(256);
    hipLaunchKernelGGL(SoftRasterizer_wmma_kernel, grid, block, 0, stream,
                       poly, masks, out);
}